// MAFM_68779606279022
// MI455X (gfx1250) — compile-verified
//
#include <hip/hip_runtime.h>
#include <hip/hip_bf16.h>

typedef float v2f __attribute__((ext_vector_type(2)));
typedef float v8f __attribute__((ext_vector_type(8)));

#define LN_EPS 1e-5f
#define HALF_LOG2E 0.72134752044448170368f   // 0.5 * log2(e), folds 1/sqrt(E)

// Raw IEEE maximumNumber without clang's sNaN canonicalization ops.
__device__ __forceinline__ float vmaxf(float a, float b) {
    float d;
    asm("v_max_num_f32 %0, %1, %2" : "=v"(d) : "v"(a), "v"(b));
    return d;
}

// ---- DPP ROW_XMASK cross-lane helpers (reduce within each 16-lane half) ----
template <int CTRL>
__device__ __forceinline__ float dpp_xor_f32(float x) {
    int r = __builtin_amdgcn_update_dpp(0, __float_as_int(x), CTRL, 0xf, 0xf, true);
    return __int_as_float(r);
}
__device__ __forceinline__ float rowmax16(float x) {
    x = vmaxf(x, dpp_xor_f32<0x161>(x));
    x = vmaxf(x, dpp_xor_f32<0x162>(x));
    x = vmaxf(x, dpp_xor_f32<0x164>(x));
    x = vmaxf(x, dpp_xor_f32<0x168>(x));
    return x;
}
__device__ __forceinline__ float rowsum16(float x) {
    x += dpp_xor_f32<0x161>(x);
    x += dpp_xor_f32<0x162>(x);
    x += dpp_xor_f32<0x164>(x);
    x += dpp_xor_f32<0x168>(x);
    return x;
}

// Fused post-norm TransformerEncoderLayer (nhead=1, E=4, FF=16), flash-style
// two-pass attention with WMMA f32 16x16x4 score tiles. One WG per batch.
template <int S>
__global__ __launch_bounds__(256) void encoder_kernel(
    const float* __restrict__ in_scalar,                                  // [B,S]
    const float* __restrict__ w_in,  const float* __restrict__ b_in,      // (1,4),(4)
    const float* __restrict__ qkv_w, const float* __restrict__ qkv_b,     // (4,12),(12)
    const float* __restrict__ o_w,   const float* __restrict__ o_b,       // (4,4),(4)
    const float* __restrict__ ln1_g, const float* __restrict__ ln1_b,
    const float* __restrict__ ff1_w, const float* __restrict__ ff1_b,     // (4,16),(16)
    const float* __restrict__ ff2_w, const float* __restrict__ ff2_b,     // (16,4),(4)
    const float* __restrict__ ln2_g, const float* __restrict__ ln2_b,
    const float* __restrict__ w_out, const float* __restrict__ b_out,     // (4,1),(1)
    float* __restrict__ out_vec)                                          // [B,S]
{
    __shared__ float sx[S * 4];   // x = token scalar @ w_in + b_in
    __shared__ float sk[S * 4];   // K projection
    __shared__ float sv[S * 4];   // V projection

    const int b   = blockIdx.x;
    const int tid = threadIdx.x;

    // ---- Phase 1: input projection + K/V into LDS --------------------------
    for (int s = tid; s < S; s += 256) {
        float t  = in_scalar[b * S + s];
        float x0 = fmaf(t, w_in[0], b_in[0]);
        float x1 = fmaf(t, w_in[1], b_in[1]);
        float x2 = fmaf(t, w_in[2], b_in[2]);
        float x3 = fmaf(t, w_in[3], b_in[3]);
        sx[s * 4 + 0] = x0; sx[s * 4 + 1] = x1;
        sx[s * 4 + 2] = x2; sx[s * 4 + 3] = x3;
#pragma unroll
        for (int e = 0; e < 4; ++e) {
            float kk = qkv_b[4 + e];
            kk = fmaf(x0, qkv_w[0 * 12 + 4 + e], kk);
            kk = fmaf(x1, qkv_w[1 * 12 + 4 + e], kk);
            kk = fmaf(x2, qkv_w[2 * 12 + 4 + e], kk);
            kk = fmaf(x3, qkv_w[3 * 12 + 4 + e], kk);
            sk[s * 4 + e] = kk;
            float vv = qkv_b[8 + e];
            vv = fmaf(x0, qkv_w[0 * 12 + 8 + e], vv);
            vv = fmaf(x1, qkv_w[1 * 12 + 8 + e], vv);
            vv = fmaf(x2, qkv_w[2 * 12 + 8 + e], vv);
            vv = fmaf(x3, qkv_w[3 * 12 + 8 + e], vv);
            sv[s * 4 + e] = vv;
        }
    }
    __syncthreads();

    // ---- Phase 2: attention + fused per-token tail -------------------------
    const int wave  = tid >> 5;
    const int lane  = tid & 31;
    const int h     = lane & 15;     // position within 16-lane half
    const bool up   = lane >= 16;
    const int k0    = up ? 2 : 0;    // A/B operand K slots: {0,1} or {2,3}
    constexpr int NT = S / 16;       // row tiles / key chunks (power of two)

    for (int tile = wave; tile < NT; tile += 8) {
        // A operand: q row M = tile*16 + h, components k0/k0+1 (Q proj on the fly)
        const int rowA = tile * 16 + h;
        float xa0 = sx[rowA * 4 + 0], xa1 = sx[rowA * 4 + 1];
        float xa2 = sx[rowA * 4 + 2], xa3 = sx[rowA * 4 + 3];
        v2f A;
        A.x = qkv_b[k0];
        A.x = fmaf(xa0, qkv_w[0 * 12 + k0], A.x);
        A.x = fmaf(xa1, qkv_w[1 * 12 + k0], A.x);
        A.x = fmaf(xa2, qkv_w[2 * 12 + k0], A.x);
        A.x = fmaf(xa3, qkv_w[3 * 12 + k0], A.x);
        A.y = qkv_b[k0 + 1];
        A.y = fmaf(xa0, qkv_w[0 * 12 + k0 + 1], A.y);
        A.y = fmaf(xa1, qkv_w[1 * 12 + k0 + 1], A.y);
        A.y = fmaf(xa2, qkv_w[2 * 12 + k0 + 1], A.y);
        A.y = fmaf(xa3, qkv_w[3 * 12 + k0 + 1], A.y);

        // ---- Pass 1: lane-local running max (software-pipelined B loads) ----
        float mx[8];
#pragma unroll
        for (int r = 0; r < 8; ++r) mx[r] = -1e30f;

        float2 bk = *(const float2*)&sk[h * 4 + k0];   // prefetch chunk 0
#pragma unroll 1
        for (int ch = 0; ch < NT; ++ch) {
            const int nkey = (((ch + 1) & (NT - 1)) * 16 + h) * 4 + k0;
            const float2 nbk = *(const float2*)&sk[nkey];   // prefetch next
            v2f Bm; Bm.x = bk.x; Bm.y = bk.y;
            v8f C;
#pragma unroll
            for (int r = 0; r < 8; ++r) C[r] = 0.f;
            C = __builtin_amdgcn_wmma_f32_16x16x4_f32(
                    false, A, false, Bm, (short)0, C, false, false);
#pragma unroll
            for (int r = 0; r < 8; ++r) mx[r] = vmaxf(mx[r], C[r]);
            bk = nbk;
        }
        // One cross-lane max per tile -> exact row max; pre-scale by 0.5*log2e
        // so pass 2 uses native exp2 with no extra multiply.
#pragma unroll
        for (int r = 0; r < 8; ++r) mx[r] = rowmax16(mx[r]) * HALF_LOG2E;

        // ---- Pass 2: recompute scores, accumulate exp-sum and p@v ----------
        float l[8];
        v2f acc01[8], acc23[8];   // packed accumulators -> v_pk_fma_f32
#pragma unroll
        for (int r = 0; r < 8; ++r) {
            l[r] = 0.f;
            acc01[r].x = acc01[r].y = 0.f;
            acc23[r].x = acc23[r].y = 0.f;
        }

        bk = *(const float2*)&sk[h * 4 + k0];
        float4 vv = *(const float4*)&sv[h * 4];
#pragma unroll 1
        for (int ch = 0; ch < NT; ++ch) {
            const int nkey = ((ch + 1) & (NT - 1)) * 16 + h;
            const float2 nbk = *(const float2*)&sk[nkey * 4 + k0];
            const float4 nvv = *(const float4*)&sv[nkey * 4];
            v2f Bm; Bm.x = bk.x; Bm.y = bk.y;
            v8f C;
#pragma unroll
            for (int r = 0; r < 8; ++r) C[r] = 0.f;
            C = __builtin_amdgcn_wmma_f32_16x16x4_f32(
                    false, A, false, Bm, (short)0, C, false, false);

            v2f v01; v01.x = vv.x; v01.y = vv.y;
            v2f v23; v23.x = vv.z; v23.y = vv.w;
#pragma unroll
            for (int r = 0; r < 8; ++r) {
                // p = exp((s - m)/sqrt(E)) as exp2(fma(s, 0.5*log2e, -m'))
                float p = __builtin_amdgcn_exp2f(fmaf(C[r], HALF_LOG2E, -mx[r]));
                l[r] += p;
                v2f pp; pp.x = p; pp.y = p;
                acc01[r] = pp * v01 + acc01[r];
                acc23[r] = pp * v23 + acc23[r];
            }
            bk = nbk;
            vv = nvv;
        }

        // One cross-lane sum per tile (DPP, VALU-rate).
#pragma unroll
        for (int r = 0; r < 8; ++r) {
            l[r] = rowsum16(l[r]);
            acc01[r].x = rowsum16(acc01[r].x);
            acc01[r].y = rowsum16(acc01[r].y);
            acc23[r].x = rowsum16(acc23[r].x);
            acc23[r].y = rowsum16(acc23[r].y);
        }

        // Fused per-token tail: lanes h<8 each own one row of this half-tile.
        if (h < 8) {
            const int r   = h;
            const int row = tile * 16 + h + (up ? 8 : 0);
            float inv = 1.0f / l[r];
            float o0 = acc01[r].x * inv, o1 = acc01[r].y * inv;
            float o2 = acc23[r].x * inv, o3 = acc23[r].y * inv;
            // attn @ o_w + o_b
            float p0 = o_b[0], p1 = o_b[1], p2 = o_b[2], p3 = o_b[3];
            p0 = fmaf(o0, o_w[0], fmaf(o1, o_w[4], fmaf(o2, o_w[8],  fmaf(o3, o_w[12], p0))));
            p1 = fmaf(o0, o_w[1], fmaf(o1, o_w[5], fmaf(o2, o_w[9],  fmaf(o3, o_w[13], p1))));
            p2 = fmaf(o0, o_w[2], fmaf(o1, o_w[6], fmaf(o2, o_w[10], fmaf(o3, o_w[14], p2))));
            p3 = fmaf(o0, o_w[3], fmaf(o1, o_w[7], fmaf(o2, o_w[11], fmaf(o3, o_w[15], p3))));
            // residual + LN1
            float y0 = sx[row * 4 + 0] + p0, y1 = sx[row * 4 + 1] + p1;
            float y2 = sx[row * 4 + 2] + p2, y3 = sx[row * 4 + 3] + p3;
            float mn = 0.25f * (y0 + y1 + y2 + y3);
            float d0 = y0 - mn, d1 = y1 - mn, d2 = y2 - mn, d3 = y3 - mn;
            float var = 0.25f * (d0 * d0 + d1 * d1 + d2 * d2 + d3 * d3);
            float rs  = rsqrtf(var + LN_EPS);
            y0 = fmaf(d0 * rs, ln1_g[0], ln1_b[0]);
            y1 = fmaf(d1 * rs, ln1_g[1], ln1_b[1]);
            y2 = fmaf(d2 * rs, ln1_g[2], ln1_b[2]);
            y3 = fmaf(d3 * rs, ln1_g[3], ln1_b[3]);
            // FF: relu(y @ ff1 + b) @ ff2 + b
            float z0 = ff2_b[0], z1 = ff2_b[1], z2 = ff2_b[2], z3 = ff2_b[3];
#pragma unroll
            for (int f = 0; f < 16; ++f) {
                float hh = ff1_b[f];
                hh = fmaf(y0, ff1_w[0 * 16 + f], hh);
                hh = fmaf(y1, ff1_w[1 * 16 + f], hh);
                hh = fmaf(y2, ff1_w[2 * 16 + f], hh);
                hh = fmaf(y3, ff1_w[3 * 16 + f], hh);
                hh = fmaxf(hh, 0.0f);
                z0 = fmaf(hh, ff2_w[f * 4 + 0], z0);
                z1 = fmaf(hh, ff2_w[f * 4 + 1], z1);
                z2 = fmaf(hh, ff2_w[f * 4 + 2], z2);
                z3 = fmaf(hh, ff2_w[f * 4 + 3], z3);
            }
            // residual + LN2
            float u0 = y0 + z0, u1 = y1 + z1, u2 = y2 + z2, u3 = y3 + z3;
            mn = 0.25f * (u0 + u1 + u2 + u3);
            d0 = u0 - mn; d1 = u1 - mn; d2 = u2 - mn; d3 = u3 - mn;
            var = 0.25f * (d0 * d0 + d1 * d1 + d2 * d2 + d3 * d3);
            rs  = rsqrtf(var + LN_EPS);
            u0 = fmaf(d0 * rs, ln2_g[0], ln2_b[0]);
            u1 = fmaf(d1 * rs, ln2_g[1], ln2_b[1]);
            u2 = fmaf(d2 * rs, ln2_g[2], ln2_b[2]);
            u3 = fmaf(d3 * rs, ln2_g[3], ln2_b[3]);
            // final scalar projection
            float res = b_out[0];
            res = fmaf(u0, w_out[0], res);
            res = fmaf(u1, w_out[1], res);
            res = fmaf(u2, w_out[2], res);
            res = fmaf(u3, w_out[3], res);
            out_vec[b * S + row] = res;
        }
    }
}

// Classifier: [B,1088] @ w_cls[1088,3] + b_cls, softmax. One thread per batch row.
__global__ __launch_bounds__(64) void cls_kernel(
    const float* __restrict__ img_vec,   // [64,1024]
    const float* __restrict__ ind_vec,   // [64,64]
    const float* __restrict__ w_cls,     // [1088,3]
    const float* __restrict__ b_cls,     // [3]
    float* __restrict__ out)             // [64,3]
{
    int b = threadIdx.x;
    if (b >= 64) return;
    float L0 = b_cls[0], L1 = b_cls[1], L2 = b_cls[2];
    for (int t = 0; t < 1024; ++t) {
        float x = img_vec[b * 1024 + t];
        L0 = fmaf(x, w_cls[t * 3 + 0], L0);
        L1 = fmaf(x, w_cls[t * 3 + 1], L1);
        L2 = fmaf(x, w_cls[t * 3 + 2], L2);
    }
    for (int t = 0; t < 64; ++t) {
        float x = ind_vec[b * 64 + t];
        int w = (1024 + t) * 3;
        L0 = fmaf(x, w_cls[w + 0], L0);
        L1 = fmaf(x, w_cls[w + 1], L1);
        L2 = fmaf(x, w_cls[w + 2], L2);
    }
    float mx = fmaxf(L0, fmaxf(L1, L2));
    float e0 = __expf(L0 - mx), e1 = __expf(L1 - mx), e2 = __expf(L2 - mx);
    float inv = 1.0f / (e0 + e1 + e2);
    out[b * 3 + 0] = e0 * inv;
    out[b * 3 + 1] = e1 * inv;
    out[b * 3 + 2] = e2 * inv;
}

extern "C" void kernel_launch(void* const* d_in, const int* in_sizes, int n_in,
                              void* d_out, int out_size, void* d_ws, size_t ws_size,
                              hipStream_t stream)
{
    const float* image  = (const float*)d_in[0];   // [64,1024,1]
    const float* indic  = (const float*)d_in[1];   // [64,1,64]
    const float* w_in1  = (const float*)d_in[2];
    const float* b_in1  = (const float*)d_in[3];
    const float* w_in2  = (const float*)d_in[4];
    const float* b_in2  = (const float*)d_in[5];
    const float* w_out1 = (const float*)d_in[6];
    const float* b_out1 = (const float*)d_in[7];
    const float* w_out2 = (const float*)d_in[8];
    const float* b_out2 = (const float*)d_in[9];
    const float* t1[12];
    const float* t2[12];
    for (int i = 0; i < 12; ++i) t1[i] = (const float*)d_in[10 + i];
    for (int i = 0; i < 12; ++i) t2[i] = (const float*)d_in[22 + i];
    const float* w_cls = (const float*)d_in[34];
    const float* b_cls = (const float*)d_in[35];

    float* img_vec = (float*)d_ws;            // 64*1024 f32
    float* ind_vec = img_vec + 64 * 1024;     // 64*64 f32

    encoder_kernel<1024><<<64, 256, 0, stream>>>(
        image, w_in1, b_in1,
        t1[0], t1[1], t1[2], t1[3], t1[4], t1[5],
        t1[6], t1[7], t1[8], t1[9], t1[10], t1[11],
        w_out1, b_out1, img_vec);

    encoder_kernel<64><<<64, 256, 0, stream>>>(
        indic, w_in2, b_in2,
        t2[0], t2[1], t2[2], t2[3], t2[4], t2[5],
        t2[6], t2[7], t2[8], t2[9], t2[10], t2[11],
        w_out2, b_out2, ind_vec);

    cls_kernel<<<1, 64, 0, stream>>>(img_vec, ind_vec, w_cls, b_cls, (float*)d_out);
}